// NodeBlock_21852793602131
// MI455X (gfx1250) — compile-verified
//
#include <hip/hip_runtime.h>

typedef __attribute__((ext_vector_type(2))) float v2f;
typedef __attribute__((ext_vector_type(8))) float v8f;

#define NN 100000
#define NE 1600000

// D = A(16x4 f32) * B(4x16 f32) + C(16x16 f32)
__device__ __forceinline__ v8f wmma4(v2f a, v2f b, v8f c) {
  return __builtin_amdgcn_wmma_f32_16x16x4_f32(
      /*neg_a=*/false, a, /*neg_b=*/false, b,
      /*c_mod=*/(short)0, c, /*reuse_a=*/false, /*reuse_b=*/false);
}

__global__ __launch_bounds__(256) void zero_ws_kernel(float4* __restrict__ p, int n4) {
  int i = blockIdx.x * 256 + threadIdx.x;
  if (i < n4) p[i] = make_float4(0.f, 0.f, 0.f, 0.f);
}

// One wave per edge row: lane = feature column. Streaming (non-temporal)
// 128B loads; f32 atomics land in the 12.8MB agg buffer, which stays hot in
// the 192MB L2 because the 218MB edge stream bypasses with TH=NT.
__global__ __launch_bounds__(256) void edge_scatter_kernel(
    const float* __restrict__ edge_attr,
    const long long* __restrict__ receivers,   // edge_index row 1
    float* __restrict__ agg) {
  int lane = threadIdx.x & 31;
  int wave = (blockIdx.x * 256 + threadIdx.x) >> 5;
  int nw   = (gridDim.x * 256) >> 5;
  for (int e = wave; e < NE; e += nw) {
    int r   = (int)__builtin_nontemporal_load(receivers + e);  // wave-uniform
    float v = __builtin_nontemporal_load(edge_attr + (size_t)e * 32 + lane);
    atomicAdd(&agg[(size_t)r * 32 + lane], v);
  }
}

// 8 waves / block, each wave produces a 16-node x 32-feature output tile
// via fp32 WMMA: out = relu([node|agg|g] @ W1 + b1) @ W2 + b2.
// Weights are stored in LDS as K-pairs (float2) so every B fragment is one
// aligned ds_load_b64 into an even VGPR pair (no repack movs before WMMA).
__global__ __launch_bounds__(256) void mlp_wmma_kernel(
    const float* __restrict__ node_attr,
    const float* __restrict__ agg,
    const float* __restrict__ gattr,
    const float* __restrict__ W1, const float* __restrict__ b1,
    const float* __restrict__ W2, const float* __restrict__ b2,
    float* __restrict__ out, int n_nodes) {
  __shared__ v2f   sW1p[48 * 32];   // pair p: (W1[2p][n], W1[2p+1][n])
  __shared__ v2f   sW2p[16 * 32];   // pair p: (W2[2p][n], W2[2p+1][n])
  __shared__ float sB1[32];
  __shared__ float sB2[32];
  __shared__ float sG[32];
  __shared__ float sH[8][16][34];   // per-wave h staging; even stride => v2f reads

  for (int i = threadIdx.x; i < 48 * 32; i += 256) {
    int p = i >> 5, n = i & 31;
    v2f w;
    w.x = W1[(2 * p)     * 32 + n];
    w.y = W1[(2 * p + 1) * 32 + n];
    sW1p[i] = w;
  }
  for (int i = threadIdx.x; i < 16 * 32; i += 256) {
    int p = i >> 5, n = i & 31;
    v2f w;
    w.x = W2[(2 * p)     * 32 + n];
    w.y = W2[(2 * p + 1) * 32 + n];
    sW2p[i] = w;
  }
  if (threadIdx.x < 32) {
    sB1[threadIdx.x] = b1[threadIdx.x];
    sB2[threadIdx.x] = b2[threadIdx.x];
    sG[threadIdx.x]  = gattr[threadIdx.x];
  }
  __syncthreads();

  const int waveId = threadIdx.x >> 5;
  const int lane   = threadIdx.x & 31;
  const int hi     = lane >> 4;      // 0: K pair {0,1} / M 0-7 ; 1: K pair {2,3} / M 8-15
  const int lo     = lane & 15;

  const int tile = blockIdx.x * 8 + waveId;
  const int m0   = tile * 16;
  if (m0 >= n_nodes) return;

  const int row  = m0 + lo;
  const int node = (row < n_nodes) ? row : (n_nodes - 1);
  const size_t nbase = (size_t)node * 32;

  v8f acc0 = {};   // output cols 0..15
  v8f acc1 = {};   // output cols 16..31

  // ----- GEMM1: K = 96 (node_attr | agg | global), 24 K-steps x 2 N-tiles -----
#pragma unroll
  for (int k0 = 0; k0 < 32; k0 += 4) {
    int p = (k0 >> 1) + hi;                       // W1 K-pair index
    v2f a  = *(const v2f*)(node_attr + nbase + k0 + 2 * hi);
    v2f bA = sW1p[p * 32 + lo];
    v2f bB = sW1p[p * 32 + 16 + lo];
    acc0 = wmma4(a, bA, acc0);
    acc1 = wmma4(a, bB, acc1);
  }
#pragma unroll
  for (int k0 = 0; k0 < 32; k0 += 4) {
    int p = 16 + (k0 >> 1) + hi;
    v2f a  = *(const v2f*)(agg + nbase + k0 + 2 * hi);
    v2f bA = sW1p[p * 32 + lo];
    v2f bB = sW1p[p * 32 + 16 + lo];
    acc0 = wmma4(a, bA, acc0);
    acc1 = wmma4(a, bB, acc1);
  }
#pragma unroll
  for (int k0 = 0; k0 < 32; k0 += 4) {
    int p = 32 + (k0 >> 1) + hi;
    v2f a  = *(const v2f*)(sG + k0 + 2 * hi);     // 8B-aligned LDS read
    v2f bA = sW1p[p * 32 + lo];
    v2f bB = sW1p[p * 32 + 16 + lo];
    acc0 = wmma4(a, bA, acc0);
    acc1 = wmma4(a, bB, acc1);
  }

  // ----- bias + ReLU, restripe D-layout (M in vgpr, N in lane) to LDS -----
#pragma unroll
  for (int r = 0; r < 8; ++r) {
    float v0 = fmaxf(acc0[r] + sB1[lo],      0.f);
    float v1 = fmaxf(acc1[r] + sB1[16 + lo], 0.f);
    sH[waveId][r + 8 * hi][lo]      = v0;
    sH[waveId][r + 8 * hi][16 + lo] = v1;
  }

  // ----- GEMM2: K = 32, A re-read in A-layout (M in lane, K pair in vgprs) -----
  v8f o0 = {};
  v8f o1 = {};
#pragma unroll
  for (int k0 = 0; k0 < 32; k0 += 4) {
    int p = (k0 >> 1) + hi;                       // W2 K-pair index
    v2f a  = *(const v2f*)&sH[waveId][lo][k0 + 2 * hi];  // even stride => aligned
    v2f bA = sW2p[p * 32 + lo];
    v2f bB = sW2p[p * 32 + 16 + lo];
    o0 = wmma4(a, bA, o0);
    o1 = wmma4(a, bB, o1);
  }

  // ----- bias + store -----
#pragma unroll
  for (int r = 0; r < 8; ++r) {
    int m = m0 + r + 8 * hi;
    if (m < n_nodes) {
      out[(size_t)m * 32 + lo]      = o0[r] + sB2[lo];
      out[(size_t)m * 32 + 16 + lo] = o1[r] + sB2[16 + lo];
    }
  }
}

extern "C" void kernel_launch(void* const* d_in, const int* in_sizes, int n_in,
                              void* d_out, int out_size, void* d_ws, size_t ws_size,
                              hipStream_t stream) {
  const float*     node_attr   = (const float*)d_in[0];
  const long long* edge_index  = (const long long*)d_in[1];  // int64, shape (2, NE)
  const float*     edge_attr   = (const float*)d_in[2];
  const float*     global_attr = (const float*)d_in[3];
  const float*     W1          = (const float*)d_in[4];
  const float*     b1          = (const float*)d_in[5];
  const float*     W2          = (const float*)d_in[6];
  const float*     b2          = (const float*)d_in[7];
  float*           out         = (float*)d_out;
  float*           agg         = (float*)d_ws;               // NN*32 f32 = 12.8 MB

  // 1) zero the aggregation buffer
  {
    int n4 = (NN * 32) / 4;
    zero_ws_kernel<<<(n4 + 255) / 256, 256, 0, stream>>>((float4*)agg, n4);
  }
  // 2) bandwidth-bound edge scatter (one wave per edge row, NT streaming loads)
  edge_scatter_kernel<<<4096, 256, 0, stream>>>(edge_attr, edge_index + NE, agg);
  // 3) fused WMMA MLP: 6250 16-node tiles, 8 per block
  {
    int tiles  = (NN + 15) / 16;
    int blocks = (tiles + 7) / 8;
    mlp_wmma_kernel<<<blocks, 256, 0, stream>>>(node_attr, agg, global_attr,
                                                W1, b1, W2, b2, out, NN);
  }
}